// MultiLevelHierarchicalPrototypes_53824530153590
// MI455X (gfx1250) — compile-verified
//
#include <hip/hip_runtime.h>
#include <hip/hip_bf16.h>

#define N_SUPPORT 131072
#define DIM 512
#define NCLS 1024
#define NLVL 3
#define LN_EPS 1e-5f
#define RT_PER_WAVE 4           // 16-row tiles per wave (M-blocking)
#define ROWS (RT_PER_WAVE * 16) // 64 rows per block

typedef float v2f __attribute__((ext_vector_type(2)));
typedef float v8f __attribute__((ext_vector_type(8)));

__global__ void mlhp_zero_kernel(float* __restrict__ p, int n) {
    int i = blockIdx.x * blockDim.x + threadIdx.x;
    if (i < n) p[i] = 0.0f;
}

__global__ void mlhp_count_kernel(const int* __restrict__ labels, float* __restrict__ counts) {
    int i = blockIdx.x * blockDim.x + threadIdx.x;
    if (i < N_SUPPORT) unsafeAtomicAdd(&counts[labels[i]], 1.0f);
}

__global__ void mlhp_softmax3_kernel(const float* __restrict__ t, float* __restrict__ w) {
    if (threadIdx.x == 0) {
        float m = fmaxf(t[0], fmaxf(t[1], t[2]));
        float e0 = __expf(t[0] - m);
        float e1 = __expf(t[1] - m);
        float e2 = __expf(t[2] - m);
        float s = e0 + e1 + e2;
        w[0] = e0 / s; w[1] = e1 / s; w[2] = e2 / s;
    }
}

// One block = 64 support rows x full D=512, one level. 256 threads = 8 wave32 waves.
// Wave w owns column tiles [4w, 4w+4) and all four 16-row tiles -> 16 accumulator
// tiles (128 VGPRs); each B fragment load feeds 4 WMMAs.
__launch_bounds__(256, 1)
__global__ void mlhp_proto_kernel(const float* __restrict__ X,
                                  const int* __restrict__ labels,
                                  const float* __restrict__ W1a,
                                  const float* __restrict__ b1a,
                                  const float* __restrict__ ga,
                                  const float* __restrict__ ba,
                                  const float* __restrict__ W2a,
                                  const float* __restrict__ b2a,
                                  const float* __restrict__ counts,
                                  const float* __restrict__ weights,
                                  float* __restrict__ out) {
    constexpr int LDSW = DIM + 8;           // stride 520 floats -> LDS bank stride 8
    __shared__ float hs[ROWS][LDSW];        // 133,120 B (< 320 KB WGP LDS)

    const int level   = blockIdx.y;
    const int rowBase = blockIdx.x * ROWS;
    const int lane = threadIdx.x & 31;      // wave32
    const int wave = threadIdx.x >> 5;      // 0..7
    const int m  = lane & 15;               // row (A) / col (B,C) index within 16
    const int kh = lane >> 4;               // 0/1: K-pair select (A/B), row-half (C/D)

    const float* __restrict__ W1 = W1a + (size_t)level * DIM * DIM;
    const float* __restrict__ W2 = W2a + (size_t)level * DIM * DIM;
    const float* __restrict__ b1 = b1a + level * DIM;
    const float* __restrict__ g  = ga  + level * DIM;
    const float* __restrict__ bt = ba  + level * DIM;
    const float* __restrict__ b2 = b2a + level * DIM;

    // ---------------- GEMM1: h = X_tile @ W1 (f32 WMMA 16x16x4) ----------------
    v8f acc[RT_PER_WAVE][4] = {};
    const float* __restrict__ xrow[RT_PER_WAVE];
#pragma unroll
    for (int rt = 0; rt < RT_PER_WAVE; ++rt)
        xrow[rt] = X + (size_t)(rowBase + rt * 16 + m) * DIM;

    for (int k0 = 0; k0 < DIM; k0 += 4) {
        const int kb = k0 + 2 * kh;                       // even -> 8B aligned
        v2f a[RT_PER_WAVE];
#pragma unroll
        for (int rt = 0; rt < RT_PER_WAVE; ++rt)
            a[rt] = *(const v2f*)(xrow[rt] + kb);         // A: M=m, K={kb,kb+1}
#pragma unroll
        for (int ci = 0; ci < 4; ++ci) {
            const int n = (wave * 4 + ci) * 16 + m;
            v2f b;
            b.x = W1[(size_t)kb * DIM + n];               // B: K=kb,   N=n
            b.y = W1[(size_t)(kb + 1) * DIM + n];         // B: K=kb+1, N=n
#pragma unroll
            for (int rt = 0; rt < RT_PER_WAVE; ++rt)
                acc[rt][ci] = __builtin_amdgcn_wmma_f32_16x16x4_f32(
                    false, a[rt], false, b, (short)0, acc[rt][ci], false, false);
        }
    }

    // Spill h (+b1) to LDS. C/D layout: VGPR r, lane -> (M = r + 8*kh, N = m+tile).
#pragma unroll
    for (int ci = 0; ci < 4; ++ci) {
        const int n = (wave * 4 + ci) * 16 + m;
        const float bb = b1[n];
#pragma unroll
        for (int rt = 0; rt < RT_PER_WAVE; ++rt) {
#pragma unroll
            for (int r = 0; r < 8; ++r) {
                hs[rt * 16 + r + 8 * kh][n] = acc[rt][ci][r] + bb;
            }
        }
    }
    __syncthreads();

    // ---------------- LayerNorm + ReLU (8 rows per wave, wave32 reduce) ----------------
#pragma unroll
    for (int rr = 0; rr < ROWS / 8; ++rr) {
        const int row = wave * (ROWS / 8) + rr;
        float s = 0.0f, s2 = 0.0f;
        for (int c = lane; c < DIM; c += 32) {
            float v = hs[row][c];
            s += v; s2 += v * v;
        }
#pragma unroll
        for (int off = 16; off > 0; off >>= 1) {
            s  += __shfl_xor(s,  off, 32);
            s2 += __shfl_xor(s2, off, 32);
        }
        const float mu   = s * (1.0f / DIM);
        const float var  = s2 * (1.0f / DIM) - mu * mu;
        const float rstd = rsqrtf(var + LN_EPS);
        for (int c = lane; c < DIM; c += 32) {
            float v = (hs[row][c] - mu) * rstd * g[c] + bt[c];
            hs[row][c] = fmaxf(v, 0.0f);
        }
    }
    __syncthreads();

    // ---------------- GEMM2: h2 = hs @ W2 (A fragments from LDS) ----------------
    v8f acc2[RT_PER_WAVE][4] = {};
    for (int k0 = 0; k0 < DIM; k0 += 4) {
        const int kb = k0 + 2 * kh;
        v2f a[RT_PER_WAVE];
#pragma unroll
        for (int rt = 0; rt < RT_PER_WAVE; ++rt)
            a[rt] = *(const v2f*)(&hs[rt * 16 + m][kb]);  // ds_load_b64
#pragma unroll
        for (int ci = 0; ci < 4; ++ci) {
            const int n = (wave * 4 + ci) * 16 + m;
            v2f b;
            b.x = W2[(size_t)kb * DIM + n];
            b.y = W2[(size_t)(kb + 1) * DIM + n];
#pragma unroll
            for (int rt = 0; rt < RT_PER_WAVE; ++rt)
                acc2[rt][ci] = __builtin_amdgcn_wmma_f32_16x16x4_f32(
                    false, a[rt], false, b, (short)0, acc2[rt][ci], false, false);
        }
    }

    // ---------------- epilogue: out[label] += w_l * (h2 + b2) / count ----------------
    // Process one row-tile at a time to keep label/scale register pressure low.
    const float wlev = weights[level];
#pragma unroll
    for (int rt = 0; rt < RT_PER_WAVE; ++rt) {
        float scale[8];
        int   lab[8];
#pragma unroll
        for (int r = 0; r < 8; ++r) {
            const int grow = rowBase + rt * 16 + r + 8 * kh;
            lab[r]   = labels[grow];
            scale[r] = wlev / fmaxf(counts[lab[r]], 1.0f);
        }
#pragma unroll
        for (int ci = 0; ci < 4; ++ci) {
            const int n = (wave * 4 + ci) * 16 + m;
            const float bb = b2[n];
#pragma unroll
            for (int r = 0; r < 8; ++r) {
                const float v = (acc2[rt][ci][r] + bb) * scale[r];
                unsafeAtomicAdd(&out[(size_t)lab[r] * DIM + n], v);
            }
        }
    }
}

extern "C" void kernel_launch(void* const* d_in, const int* in_sizes, int n_in,
                              void* d_out, int out_size, void* d_ws, size_t ws_size,
                              hipStream_t stream) {
    const float* X      = (const float*)d_in[0];
    const int*   labels = (const int*)  d_in[1];
    const float* W1     = (const float*)d_in[2];
    const float* b1     = (const float*)d_in[3];
    const float* gamma  = (const float*)d_in[4];
    const float* beta   = (const float*)d_in[5];
    const float* W2     = (const float*)d_in[6];
    const float* b2     = (const float*)d_in[7];
    const float* temps  = (const float*)d_in[8];
    float* out = (float*)d_out;

    float* counts  = (float*)d_ws;      // [1024]
    float* weights = counts + NCLS;     // [3]

    // zero atomic targets
    mlhp_zero_kernel<<<(NCLS * DIM + 255) / 256, 256, 0, stream>>>(out, NCLS * DIM);
    mlhp_zero_kernel<<<(NCLS + 255) / 256, 256, 0, stream>>>(counts, NCLS);

    // class counts + level softmax weights
    mlhp_count_kernel<<<N_SUPPORT / 256, 256, 0, stream>>>(labels, counts);
    mlhp_softmax3_kernel<<<1, 32, 0, stream>>>(temps, weights);

    // fused MLP + segment-mean + level-weighted accumulate
    dim3 grid(N_SUPPORT / ROWS, NLVL);
    mlhp_proto_kernel<<<grid, 256, 0, stream>>>(X, labels, W1, b1, gamma, beta,
                                                W2, b2, counts, weights, out);
}